// SphericalResample_69131793596428
// MI455X (gfx1250) — compile-verified
//
#include <hip/hip_runtime.h>
#include <hip/hip_bf16.h>
#include <stdint.h>

// Spherical resample: out[b,p,k,c] = sum_j x[b, idx[p,k,j], c] * w[p,k,j]
// B=32, NLAT*NLON=8192 points, 9 taps, C=32 channels.
// Bandwidth-bound (~340 MB HBM, ~15us floor @ 23.3 TB/s). x (33.5 MB) is kept
// L2-resident; output is streamed with non-temporal stores; interpolation
// tables are staged into LDS with CDNA5 async global->LDS copies.

typedef float v4f __attribute__((ext_vector_type(4)));
typedef int   v4i __attribute__((ext_vector_type(4)));

#define NPOINTS   8192      // NLAT*NLON
#define NTAPS     9
#define NCH       32
#define NBATCH    32
#define NPK       (NPOINTS * NTAPS)   // 73728
#define PTS       32                  // (p,k) pairs per block

__global__ __launch_bounds__(256)
void SphericalResample_gather_kernel(const float* __restrict__ x,
                                     const int*   __restrict__ idx,
                                     const float* __restrict__ w,
                                     float*       __restrict__ out)
{
    __shared__ v4i s_idx[PTS];   // 512 B
    __shared__ v4f s_w[PTS];     // 512 B

    const int t   = threadIdx.x;
    const int b   = t >> 3;          // 0..31 : batch
    const int c   = (t & 7) << 2;    // 0,4,...,28 : channel quad
    const int pk0 = blockIdx.x * PTS;

    // ---- Stage per-block idx/w tables into LDS via CDNA5 async copy ----
    // Lanes 0..31 pull idx (32 x int4 = 512B), lanes 32..63 pull w (512B).
    // Each lane issues one 128-bit async global->LDS transfer (ASYNCcnt).
    if (t < PTS) {
        uint32_t lds_off = (uint32_t)(uintptr_t)&s_idx[t];  // low 32 bits = LDS byte addr
        uint64_t gaddr   = (uint64_t)(uintptr_t)(idx + (size_t)(pk0 + t) * 4);
        asm volatile("global_load_async_to_lds_b128 %0, %1, off"
                     :: "v"(lds_off), "v"(gaddr) : "memory");
    } else if (t < 2 * PTS) {
        const int u = t - PTS;
        uint32_t lds_off = (uint32_t)(uintptr_t)&s_w[u];
        uint64_t gaddr   = (uint64_t)(uintptr_t)(w + (size_t)(pk0 + u) * 4);
        asm volatile("global_load_async_to_lds_b128 %0, %1, off"
                     :: "v"(lds_off), "v"(gaddr) : "memory");
    }
    asm volatile("s_wait_asynccnt 0x0" ::: "memory");
    __syncthreads();

    // x row base for this (batch, channel-quad); rows are 32 floats.
    const float* xb = x   + (size_t)b * (NPOINTS * NCH) + c;
    float*       ob = out + ((size_t)b * NPK + pk0) * NCH + c;

    // For fixed j, a wave32 (4 batches x 8 channel-quads) issues 4 coalesced
    // 128B segments; gathers hit L2 since x (33.5 MB) << 192 MB L2.
#pragma unroll 4
    for (int pi = 0; pi < PTS; ++pi) {
        v4i id = s_idx[pi];   // uniform per iteration -> LDS broadcast read
        v4f ww = s_w[pi];

        v4f a0 = *(const v4f*)(xb + (size_t)id.x * NCH);
        v4f a1 = *(const v4f*)(xb + (size_t)id.y * NCH);
        v4f a2 = *(const v4f*)(xb + (size_t)id.z * NCH);
        v4f a3 = *(const v4f*)(xb + (size_t)id.w * NCH);

        v4f acc = a0 * ww.x + a1 * ww.y + a2 * ww.z + a3 * ww.w;

        // Streaming 302MB output: non-temporal so it doesn't evict x from L2.
        __builtin_nontemporal_store(acc, (v4f*)(ob + (size_t)pi * NCH));
    }
}

extern "C" void kernel_launch(void* const* d_in, const int* in_sizes, int n_in,
                              void* d_out, int out_size, void* d_ws, size_t ws_size,
                              hipStream_t stream) {
    const float* x   = (const float*)d_in[0];  // (32, 64, 128, 32) f32
    const int*   idx = (const int*)  d_in[1];  // (8192, 9, 4) i32
    const float* w   = (const float*)d_in[2];  // (8192, 9, 4) f32
    float*       out = (float*)d_out;          // (32, 8192, 9, 32) f32

    dim3 grid(NPK / PTS);   // 2304 blocks
    dim3 block(256);        // 8 wave32s: 32 batches x 8 channel-quads
    SphericalResample_gather_kernel<<<grid, block, 0, stream>>>(x, idx, w, out);
}